// CoraGCN_71571335020794
// MI455X (gfx1250) — compile-verified
//
#include <hip/hip_runtime.h>
#include <cstdint>
#include <cstddef>

#define N_NODES 50000
#define N_EDGES 800000
#define IN_F    128
#define HID     128
#define N_CLS   40
#define N_CLS_P 48      // N_CLS padded to a multiple of 16 for guard-free WMMA

typedef float v2f __attribute__((ext_vector_type(2)));
typedef float v8f __attribute__((ext_vector_type(8)));

// ---------------- degree / normalization precompute ----------------

__global__ __launch_bounds__(256) void zero_f32(float* __restrict__ p, int n) {
    int i = blockIdx.x * blockDim.x + threadIdx.x;
    if (i < n) p[i] = 0.0f;
}

__global__ __launch_bounds__(256) void deg_accum(const long long* __restrict__ col,
                                                 const float* __restrict__ ew,
                                                 float* __restrict__ deg, int E) {
    int e = blockIdx.x * blockDim.x + threadIdx.x;
    if (e < E) atomicAdd(&deg[(int)col[e]], ew[e]);
}

__global__ __launch_bounds__(256) void deg_to_dinv(float* __restrict__ d, int n) {
    int i = blockIdx.x * blockDim.x + threadIdx.x;
    if (i < n) {
        float v = d[i] + 1.0f;                // + self-loop weight 1
        d[i] = (v > 0.0f) ? rsqrtf(v) : 0.0f; // deg >= 1 always, guard anyway
    }
}

__global__ __launch_bounds__(256) void edge_norm(const long long* __restrict__ row,
                                                 const long long* __restrict__ col,
                                                 const float* __restrict__ ew,
                                                 const float* __restrict__ dinv,
                                                 float* __restrict__ norm, int E) {
    int e = blockIdx.x * blockDim.x + threadIdx.x;
    if (e < E) norm[e] = dinv[(int)row[e]] * ew[e] * dinv[(int)col[e]];
}

// Zero-pad W3 [K x N] into Wp [K x Np]
__global__ __launch_bounds__(256) void pad_weight(const float* __restrict__ W,
                                                  float* __restrict__ Wp,
                                                  int K, int N, int Np) {
    int idx = blockIdx.x * blockDim.x + threadIdx.x;
    if (idx >= K * Np) return;
    int k = idx / Np;
    int n = idx - k * Np;
    Wp[idx] = (n < N) ? W[k * N + n] : 0.0f;
}

// ---------------- dense GEMM via V_WMMA_F32_16X16X4_F32 ----------------
// C[M,N] = (RELU ? relu(A) : A)[M,K] @ B[K,N], row-major.
// M % 16 == 0, N % 16 == 0 (weights pre-padded), K compile-time.
// One wave computes one 16x16 tile; 8 waves (256 threads) per block.
// Inner loop is branch-free: EXEC stays all-ones around every WMMA.

template <int K, bool RELU>
__global__ __launch_bounds__(256)
void gemm_wmma(const float* __restrict__ A, const float* __restrict__ B,
               float* __restrict__ C, int M, int N) {
    const int lane    = threadIdx.x & 31;
    const int wave    = threadIdx.x >> 5;
    const int tiles_n = N >> 4;
    const int tiles_m = M >> 4;
    const int tile    = blockIdx.x * 8 + wave;
    if (tile >= tiles_m * tiles_n) return;      // wave-uniform exit
    const int tm = tile / tiles_n;
    const int tn = tile - tm * tiles_n;

    const int mrow  = (tm << 4) + (lane & 15);  // A-operand row for this lane
    const int ncol  = (tn << 4) + (lane & 15);  // B-operand col for this lane
    const int khalf = (lane >> 4) << 1;         // 0 for lanes 0-15, 2 for 16-31

    const float* arow = A + (size_t)mrow * K + khalf;
    const float* bcol = B + (size_t)khalf * N + ncol;
    v8f acc = {0.f, 0.f, 0.f, 0.f, 0.f, 0.f, 0.f, 0.f};

#pragma unroll 8
    for (int k0 = 0; k0 < K; k0 += 4) {
        v2f a = *reinterpret_cast<const v2f*>(arow + k0);     // one b64 load
        if (RELU) { a[0] = fmaxf(a[0], 0.0f); a[1] = fmaxf(a[1], 0.0f); }
        v2f b;
        b[0] = bcol[(size_t)k0 * N];                          // coalesced across lanes
        b[1] = bcol[(size_t)(k0 + 1) * N];
        acc = __builtin_amdgcn_wmma_f32_16x16x4_f32(false, a, false, b,
                                                    (short)0, acc, false, false);
    }

    // C/D layout: VGPR r -> rows r (lanes 0-15) / r+8 (lanes 16-31)
    const int rbase = (tm << 4) + ((lane >> 4) << 3);
    float* crow = C + (size_t)rbase * N + (tn << 4) + (lane & 15);
#pragma unroll
    for (int r = 0; r < 8; ++r)
        crow[(size_t)r * N] = acc[r];
}

// ---------------- aggregation: out = dinv^2 * h + b, then += norm*h[row] at col ----

__global__ __launch_bounds__(256)
void agg_init(const float* __restrict__ h, const float* __restrict__ dinv,
              const float* __restrict__ bias, float* __restrict__ out,
              int n, int F, int ldh) {
    int idx = blockIdx.x * blockDim.x + threadIdx.x;
    if (idx >= n * F) return;
    int i = idx / F;
    int f = idx - i * F;
    float d = dinv[i];
    out[idx] = d * d * h[(size_t)i * ldh + f] + bias[f];   // self-loop term + bias
}

// One wave per edge; float4 gather from L2-resident h, per-float atomicAdd scatter.
__global__ __launch_bounds__(256)
void agg_edges(const long long* __restrict__ row, const long long* __restrict__ col,
               const float* __restrict__ norm, const float* __restrict__ h,
               float* __restrict__ out, int E, int F, int ldh) {
    const int lane = threadIdx.x & 31;
    const int e = blockIdx.x * (blockDim.x >> 5) + (threadIdx.x >> 5);
    if (e >= E) return;
    const int r = (int)row[e];
    const int c = (int)col[e];
    const float w = norm[e];
    const float* hr = h + (size_t)r * ldh;
    float* oc = out + (size_t)c * F;
    for (int f = lane * 4; f < F; f += 128) {
        const float4 v = *reinterpret_cast<const float4*>(hr + f);
        atomicAdd(oc + f + 0, w * v.x);
        atomicAdd(oc + f + 1, w * v.y);
        atomicAdd(oc + f + 2, w * v.z);
        atomicAdd(oc + f + 3, w * v.w);
    }
}

// ---------------- driver ----------------

extern "C" void kernel_launch(void* const* d_in, const int* in_sizes, int n_in,
                              void* d_out, int out_size, void* d_ws, size_t ws_size,
                              hipStream_t stream) {
    const float*     x   = (const float*)d_in[0];
    const float*     ew  = (const float*)d_in[1];
    const float*     W1  = (const float*)d_in[2];
    const float*     b1  = (const float*)d_in[3];
    const float*     W2  = (const float*)d_in[4];
    const float*     b2  = (const float*)d_in[5];
    const float*     W3  = (const float*)d_in[6];
    const float*     b3  = (const float*)d_in[7];
    const long long* ei  = (const long long*)d_in[8];   // int64 per reference
    const long long* row = ei;                          // edge_index[0]
    const long long* col = ei + N_EDGES;                // edge_index[1]
    float* out = (float*)d_out;                         // [50000, 40] f32

    // workspace carve-out (~64 MB of f32, all slices 16B-aligned)
    float* ws   = (float*)d_ws;
    float* dinv = ws;                               // 50016 (padded)
    float* norm = dinv + 50016;                     // 800000
    float* B1   = norm + 800000;                    // 50000*128 (h1 / h2)
    float* B2   = B1 + (size_t)N_NODES * HID;       // 50000*128 (agg1 / agg2)
    float* B3   = B2 + (size_t)N_NODES * HID;       // 50000*48  (h3, padded cols)
    float* Wp   = B3 + (size_t)N_NODES * N_CLS_P;   // 128*48    (padded W3)

    const dim3 blk(256);
    auto cdiv = [](int a, int b) { return (a + b - 1) / b; };
    auto gemm_grid = [](int M, int N) { int t = (M >> 4) * (N >> 4); return (t + 7) / 8; };

    // graph normalization (computed once, reused by all 3 layers)
    zero_f32   <<<cdiv(N_NODES, 256), blk, 0, stream>>>(dinv, N_NODES);
    deg_accum  <<<cdiv(N_EDGES, 256), blk, 0, stream>>>(col, ew, dinv, N_EDGES);
    deg_to_dinv<<<cdiv(N_NODES, 256), blk, 0, stream>>>(dinv, N_NODES);
    edge_norm  <<<cdiv(N_EDGES, 256), blk, 0, stream>>>(row, col, ew, dinv, norm, N_EDGES);
    pad_weight <<<cdiv(HID * N_CLS_P, 256), blk, 0, stream>>>(W3, Wp, HID, N_CLS, N_CLS_P);

    // layer 1: h1 = x@W1 ; a1 = A_hat h1 + b1
    gemm_wmma<IN_F, false><<<gemm_grid(N_NODES, HID), blk, 0, stream>>>(x, W1, B1, N_NODES, HID);
    agg_init <<<cdiv(N_NODES * HID, 256), blk, 0, stream>>>(B1, dinv, b1, B2, N_NODES, HID, HID);
    agg_edges<<<cdiv(N_EDGES, 8), blk, 0, stream>>>(row, col, norm, B1, B2, N_EDGES, HID, HID);

    // layer 2: h2 = relu(a1)@W2 ; a2 = A_hat h2 + b2
    gemm_wmma<HID, true><<<gemm_grid(N_NODES, HID), blk, 0, stream>>>(B2, W2, B1, N_NODES, HID);
    agg_init <<<cdiv(N_NODES * HID, 256), blk, 0, stream>>>(B1, dinv, b2, B2, N_NODES, HID, HID);
    agg_edges<<<cdiv(N_EDGES, 8), blk, 0, stream>>>(row, col, norm, B1, B2, N_EDGES, HID, HID);

    // layer 3: h3 = relu(a2)@W3p (stride 48) ; out = A_hat h3 + b3 (stride 40)
    gemm_wmma<HID, true><<<gemm_grid(N_NODES, N_CLS_P), blk, 0, stream>>>(B2, Wp, B3, N_NODES, N_CLS_P);
    agg_init <<<cdiv(N_NODES * N_CLS, 256), blk, 0, stream>>>(B3, dinv, b3, out, N_NODES, N_CLS, N_CLS_P);
    agg_edges<<<cdiv(N_EDGES, 8), blk, 0, stream>>>(row, col, norm, B3, out, N_EDGES, N_CLS, N_CLS_P);
}